// Latent_1331439862067
// MI455X (gfx1250) — compile-verified
//
#include <hip/hip_runtime.h>
#include <hip/hip_bf16.h>

// Problem constants from the reference: z[N,D], e[M,D], probs[N], dropout[N,D]
#define N_ 2048
#define M_ 512
#define D_ 128

typedef __attribute__((ext_vector_type(2))) float v2f;
typedef __attribute__((ext_vector_type(8))) float v8f;

// ---------------------------------------------------------------------------
// Workspace layout (floats):
//   [0,2048)      zn2[n]   = sum_d z[n,d]^2
//   [2048,2560)   em2[m]   = sum_d e[m,d]^2
//   [2560,3072)   minsum_bits[m] (uint bit pattern of running min, >=0 floats)
//   [3072]        wise_accum (sum over (m,d) of elementwise min)
// ---------------------------------------------------------------------------

__global__ void init_ws_kernel(unsigned int* __restrict__ minsum_bits,
                               float* __restrict__ wise_accum) {
    int i = blockIdx.x * blockDim.x + threadIdx.x;
    if (i < M_) minsum_bits[i] = 0x7F800000u;  // +inf
    if (i == 0) *wise_accum = 0.0f;
}

// One wave (32 lanes) per row; rows [0,N) are z, rows [N, N+M) are e.
// Each lane loads a float4 (row is 512B, 16B-aligned), sums of squares,
// then wave32 shuffle reduction.
__global__ void row_sqnorm_kernel(const float* __restrict__ z,
                                  const float* __restrict__ e,
                                  float* __restrict__ zn2,
                                  float* __restrict__ em2) {
    int wave = blockIdx.x * (blockDim.x >> 5) + (threadIdx.x >> 5);
    int lane = threadIdx.x & 31;
    if (wave >= N_ + M_) return;
    const float* row = (wave < N_) ? (z + (size_t)wave * D_)
                                   : (e + (size_t)(wave - N_) * D_);
    const float4 v = ((const float4*)row)[lane];   // 32 lanes * 4 = 128 = D
    float s = v.x * v.x + v.y * v.y + v.z * v.z + v.w * v.w;
    #pragma unroll
    for (int off = 16; off > 0; off >>= 1)
        s += __shfl_down(s, off, 32);
    if (lane == 0) {
        if (wave < N_) zn2[wave] = s;
        else           em2[wave - N_] = s;
    }
}

// ---------------------------------------------------------------------------
// min_sum via WMMA: dsum[n,m] = zn2[n] + em2[m] - 2 * (z @ e^T)[n,m]
// One wave per 16x16 (n,m) tile; K=128 accumulated with 32x
// V_WMMA_F32_16X16X4_F32 (fp32 in/out -> matches reference precision).
//
// A (16x4 f32) layout: lanes 0-15 row M=lane, v[0]=K0,v[1]=K1;
//                      lanes 16-31 row M=lane-16, v[0]=K2,v[1]=K3.
// B (4x16 f32) layout mirrors it over columns (B[k][m] = e[m][k], so the
// addressing pattern is identical to A with m as the "row" index).
// C/D (16x16 f32): VGPR r, lanes 0-15 -> (row=r,   col=lane),
//                          lanes16-31 -> (row=r+8, col=lane-16).
// ---------------------------------------------------------------------------
__global__ void wmma_minsum_kernel(const float* __restrict__ z,
                                   const float* __restrict__ e,
                                   const float* __restrict__ zn2,
                                   const float* __restrict__ em2,
                                   unsigned int* __restrict__ minsum_bits) {
    const int TILES_M = M_ / 16;                         // 32
    int wave = blockIdx.x * (blockDim.x >> 5) + (threadIdx.x >> 5);
    int lane = threadIdx.x & 31;
    int tn = wave / TILES_M;                             // 0..127
    int tm = wave % TILES_M;                             // 0..31
    int n0 = tn * 16, m0 = tm * 16;

    int half = lane >> 4;                                // 0 or 1
    int l16  = lane & 15;

    const float* za = z + (size_t)(n0 + l16) * D_;       // A row (n)
    const float* eb = e + (size_t)(m0 + l16) * D_;       // B col (m), row-major e

    v8f c = {};
    #pragma unroll 4
    for (int k = 0; k < D_; k += 4) {
        int kk = k + 2 * half;
        v2f a, b;
        a.x = za[kk];     a.y = za[kk + 1];
        b.x = eb[kk];     b.y = eb[kk + 1];
        c = __builtin_amdgcn_wmma_f32_16x16x4_f32(
                /*neg_a=*/false, a, /*neg_b=*/false, b,
                /*c_mod=*/(short)0, c, /*reuse_a=*/false, /*reuse_b=*/false);
    }

    // This lane owns column m0+l16 for 8 rows (r + 8*half).
    float em2v = em2[m0 + l16];
    float best = __builtin_inff();
    #pragma unroll
    for (int r = 0; r < 8; ++r) {
        int n = n0 + r + 8 * half;
        float dsum = zn2[n] + em2v - 2.0f * c[r];
        best = fminf(best, dsum);
    }
    // Merge the two row-halves living in lane l16 and lane l16+16 (same column)
    best = fminf(best, __shfl_xor(best, 16, 32));
    best = fmaxf(best, 0.0f);  // exact math is >=0; keeps uint-min ordering valid
    if (half == 0)
        atomicMin(&minsum_bits[m0 + l16], __float_as_uint(best));
}

// ---------------------------------------------------------------------------
// Elementwise min over n of (z[n,d]-e[m,d])^2, summed into wise_accum.
// Block = one m (128 lanes = D); z loads are coalesced and L2-resident.
// ---------------------------------------------------------------------------
__global__ void min_elem_kernel(const float* __restrict__ z,
                                const float* __restrict__ e,
                                float* __restrict__ wise_accum) {
    int m = blockIdx.x;
    int d = threadIdx.x;
    float ed = e[(size_t)m * D_ + d];
    float best = __builtin_inff();
    for (int n = 0; n < N_; ++n) {
        float diff = z[(size_t)n * D_ + d] - ed;
        best = fminf(best, diff * diff);
    }
    __shared__ float sdata[D_];
    sdata[d] = best;
    __syncthreads();
    #pragma unroll
    for (int s = D_ / 2; s > 0; s >>= 1) {
        if (d < s) sdata[d] += sdata[d + s];
        __syncthreads();
    }
    if (d == 0) atomicAdd(wise_accum, sdata[0]);
}

// z_out = z * (dropout < probs[n]); written to d_out[2..]
__global__ void dropout_kernel(const float* __restrict__ z,
                               const float* __restrict__ probs,
                               const float* __restrict__ dropout,
                               float* __restrict__ out) {
    int i = blockIdx.x * blockDim.x + threadIdx.x;
    if (i >= N_ * D_) return;
    int n = i / D_;
    float keep = (dropout[i] < probs[n]) ? 1.0f : 0.0f;
    out[i] = z[i] * keep;
}

// d_out[0] = mean_m(min_sum); d_out[1] = wise_accum / (M*D)
__global__ void finalize_kernel(const unsigned int* __restrict__ minsum_bits,
                                const float* __restrict__ wise_accum,
                                float* __restrict__ out) {
    __shared__ float sdata[M_];
    int t = threadIdx.x;
    sdata[t] = __uint_as_float(minsum_bits[t]);
    __syncthreads();
    #pragma unroll
    for (int s = M_ / 2; s > 0; s >>= 1) {
        if (t < s) sdata[t] += sdata[t + s];
        __syncthreads();
    }
    if (t == 0) {
        out[0] = sdata[0] / (float)M_;
        out[1] = *wise_accum / (float)(M_ * D_);
    }
}

extern "C" void kernel_launch(void* const* d_in, const int* in_sizes, int n_in,
                              void* d_out, int out_size, void* d_ws, size_t ws_size,
                              hipStream_t stream) {
    const float* z       = (const float*)d_in[0];   // [N, D]
    const float* e       = (const float*)d_in[1];   // [M, D]
    const float* probs   = (const float*)d_in[2];   // [N]
    const float* dropout = (const float*)d_in[3];   // [N, D]
    float* out = (float*)d_out;                     // [2 + N*D]

    float* ws  = (float*)d_ws;
    float* zn2 = ws;                                      // 2048
    float* em2 = ws + N_;                                 // 512
    unsigned int* msb = (unsigned int*)(ws + N_ + M_);    // 512
    float* wacc = ws + N_ + 2 * M_;                       // 1

    // 1) re-init scratch (needed every call for graph replay determinism)
    init_ws_kernel<<<(M_ + 255) / 256, 256, 0, stream>>>(msb, wacc);

    // 2) row squared norms: one wave per row, (N+M)=2560 waves, 8 waves/block
    row_sqnorm_kernel<<<(N_ + M_) / 8, 256, 0, stream>>>(z, e, zn2, em2);

    // 3) WMMA GEMM min-sum: (N/16)*(M/16)=4096 tiles, 8 waves/block -> 512 blocks
    wmma_minsum_kernel<<<(N_ / 16) * (M_ / 16) / 8, 256, 0, stream>>>(
        z, e, zn2, em2, msb);

    // 4) elementwise min over n, accumulated for the mean
    min_elem_kernel<<<M_, D_, 0, stream>>>(z, e, wacc);

    // 5) dropout-masked z -> out[2..]
    dropout_kernel<<<(N_ * D_ + 255) / 256, 256, 0, stream>>>(
        z, probs, dropout, out + 2);

    // 6) scalar reductions -> out[0], out[1]
    finalize_kernel<<<1, M_, 0, stream>>>(msb, wacc, out);
}